// DLRM_23295902614210
// MI455X (gfx1250) — compile-verified
//
#include <hip/hip_runtime.h>

typedef float v2f __attribute__((ext_vector_type(2)));
typedef float v8f __attribute__((ext_vector_type(8)));

#define B_     16384
#define F_     26
#define D_     64
#define DENSE_ 13
#define CARD_  100000
#define HBOT   8
#define HTOP   16

// One wave32 per 16-row batch tile. Fully fused DLRM:
// prefetch gather -> bottom MLP -> interaction GEMM (v_wmma_f32_16x16x4_f32)
// -> top MLP -> sigmoid.
__global__ __launch_bounds__(32) void dlrm_fused_kernel(
    const float* __restrict__ dense_x,   // [B,13]
    const int*   __restrict__ sparse_x,  // [B,26]
    const float* __restrict__ tables,    // [26,100000,64]
    const float* __restrict__ w1,        // [13,8]
    const float* __restrict__ b1,        // [8]
    const float* __restrict__ w2,        // [8,64]
    const float* __restrict__ b2,        // [64]
    const float* __restrict__ tw1,       // [1728,16]
    const float* __restrict__ tb1,       // [16]
    const float* __restrict__ tw2,       // [16,1]
    const float* __restrict__ tb2,       // [1]
    float*       __restrict__ out)       // [B,1]
{
    const int lane = threadIdx.x;       // 0..31
    const int tile = blockIdx.x;        // 0..B/16-1
    const int row  = lane & 15;         // A-fragment row M
    const int half = lane >> 4;         // 0 | 1
    const int koff = half * 2;          // per-lane K offset within a K=4 chunk
    const int r    = tile * 16 + row;   // global batch row (duplicated across halves)

    // ---- prefetch pre-pass: put all 416 gathered rows (256B each) in flight ----
    // lanes 0-15 prefetch cacheline 0 of their row, lanes 16-31 cacheline 1.
#pragma unroll
    for (int f = 0; f < F_; ++f) {
        const int pidx = sparse_x[r * F_ + f];
        const float* p = tables + ((size_t)f * CARD_ + (size_t)pidx) * D_ + half * 32;
        __builtin_prefetch(p, 0, 3);    // global_prefetch_b8
    }

    // ---------------- bottom MLP: dense_x[r,:] -> h[8] -> relu -----------------
    float h[HBOT];
#pragma unroll
    for (int i = 0; i < HBOT; ++i) h[i] = b1[i];
#pragma unroll
    for (int j = 0; j < DENSE_; ++j) {
        const float x = dense_x[r * DENSE_ + j];
#pragma unroll
        for (int i = 0; i < HBOT; ++i) h[i] = fmaf(x, w1[j * HBOT + i], h[i]);
    }
#pragma unroll
    for (int i = 0; i < HBOT; ++i) h[i] = fmaxf(h[i], 0.0f);

    // dense_out columns this lane's A-fragment slots need: c = 4*kk + koff (+1)
    v2f dval[16];
#pragma unroll
    for (int kk = 0; kk < 16; ++kk) {
        const int c0 = 4 * kk + koff;
        float s0 = b2[c0], s1 = b2[c0 + 1];
#pragma unroll
        for (int i = 0; i < HBOT; ++i) {
            s0 = fmaf(h[i], w2[i * D_ + c0],     s0);
            s1 = fmaf(h[i], w2[i * D_ + c0 + 1], s1);
        }
        dval[kk].x = s0; dval[kk].y = s1;
    }

    v8f acc = {0.f, 0.f, 0.f, 0.f, 0.f, 0.f, 0.f, 0.f};
    const float* tw1n = tw1 + row;      // column N = lane&15 of row-major [1728,16]

    // ------------- dense block of top GEMM: dense_out[16x64] @ tw1 -------------
    {
        const float* tw1d = tw1n + (F_ * D_) * HTOP;   // rows 1664..1727
#pragma unroll
        for (int kk = 0; kk < 16; ++kk) {
            v2f b;
            b.x = tw1d[(4 * kk + koff) * HTOP];
            b.y = tw1d[(4 * kk + koff + 1) * HTOP];
            acc = __builtin_amdgcn_wmma_f32_16x16x4_f32(
                false, dval[kk], false, b, (short)0, acc, false, false);
        }
    }

    // ---- 26 embedding features: gather 16 random 256B rows, 16 WMMAs each ----
    for (int f = 0; f < F_; ++f) {
        const int idx = sparse_x[r * F_ + f];          // L0-cached (prefetch pass)
        const float* src  = tables + ((size_t)f * CARD_ + (size_t)idx) * D_ + koff;
        const float* tw1f = tw1n + (f * D_) * HTOP;

        v2f a[16], b[16];
#pragma unroll
        for (int kk = 0; kk < 16; ++kk)                 // 16 independent b64 loads
            a[kk] = *(const v2f*)(src + 4 * kk);
#pragma unroll
        for (int kk = 0; kk < 16; ++kk) {               // cached tw1 reads
            b[kk].x = tw1f[(4 * kk + koff) * HTOP];
            b[kk].y = tw1f[(4 * kk + koff + 1) * HTOP];
        }
#pragma unroll
        for (int kk = 0; kk < 16; ++kk)
            acc = __builtin_amdgcn_wmma_f32_16x16x4_f32(
                false, a[kk], false, b[kk], (short)0, acc, false, false);
    }

    // -------- epilogue: relu(hidden + tb1) . tw2 + tb2, sigmoid, store --------
    // acc[i] on lane L = hidden[M = i + 8*half][N = L&15]
    const float tb1v = tb1[row];
    const float tw2v = tw2[row];
    float res = 0.0f;
#pragma unroll
    for (int i = 0; i < 8; ++i) {
        float p = fmaxf(acc[i] + tb1v, 0.0f) * tw2v;
        p += __shfl_xor(p, 1);      // reductions stay within each 16-lane half
        p += __shfl_xor(p, 2);
        p += __shfl_xor(p, 4);
        p += __shfl_xor(p, 8);
        if (row == i) res = p;      // lane (i, i+16) now holds rows i / i+8
    }
    if (row < 8) {
        const int m  = row + half * 8;
        const float z = res + tb2[0];
        out[tile * 16 + m] = 1.0f / (1.0f + __expf(-z));
    }
}

extern "C" void kernel_launch(void* const* d_in, const int* in_sizes, int n_in,
                              void* d_out, int out_size, void* d_ws, size_t ws_size,
                              hipStream_t stream) {
    (void)in_sizes; (void)n_in; (void)d_ws; (void)ws_size; (void)out_size;
    const float* dense_x  = (const float*)d_in[0];
    const int*   sparse_x = (const int*)  d_in[1];
    const float* tables   = (const float*)d_in[2];
    const float* w1       = (const float*)d_in[3];
    const float* b1       = (const float*)d_in[4];
    const float* w2       = (const float*)d_in[5];
    const float* b2       = (const float*)d_in[6];
    const float* tw1      = (const float*)d_in[7];
    const float* tb1      = (const float*)d_in[8];
    const float* tw2      = (const float*)d_in[9];
    const float* tb2      = (const float*)d_in[10];
    float* out = (float*)d_out;

    dlrm_fused_kernel<<<B_ / 16, 32, 0, stream>>>(
        dense_x, sparse_x, tables, w1, b1, w2, b2, tw1, tb1, tw2, tb2, out);
}